// DiagramNet_17763984736878
// MI455X (gfx1250) — compile-verified
//
#include <hip/hip_runtime.h>

typedef float v2f __attribute__((ext_vector_type(2)));
typedef float v8f __attribute__((ext_vector_type(8)));

#define B_   32
#define M_   128
#define N_   64
#define D_   256
#define EPSQ 1e-8f
#define NEG_BIG -9e15f

// ---------------------------------------------------------------------------
// K1: dia_norm[b,n] = ||dia[b,n,:]||  — one wave per row, 8 waves/block
// ---------------------------------------------------------------------------
__global__ void __launch_bounds__(256) k_dia_norm(const float* __restrict__ dia,
                                                  float* __restrict__ dnorm) {
    int row  = blockIdx.x * 8 + (threadIdx.x >> 5);   // [0, B_*N_)
    int lane = threadIdx.x & 31;
    const float* p = dia + (size_t)row * D_;
    float s = 0.f;
    #pragma unroll
    for (int i = 0; i < D_; i += 128) {
        float4 v = *(const float4*)(p + i + lane * 4);
        s += v.x * v.x + v.y * v.y + v.z * v.z + v.w * v.w;
    }
    #pragma unroll
    for (int off = 16; off >= 1; off >>= 1) s += __shfl_xor(s, off, 32);
    if (lane == 0) dnorm[row] = sqrtf(s);
}

// ---------------------------------------------------------------------------
// K2: avg_sim[b,m] via V_WMMA_F32_16X16X4_F32.
// One block per (b, 16-wide m tile). 8 waves split the N dimension.
// Per wave / per n:
//   B operand = dd[b, m0+lane%16, n, k + 2*(lane>=16) + {0,1}]  (b64 load)
//   A operand = dia[b, n, k + 2*(lane>=16) + {0,1}] (broadcast across M rows)
// Since all A rows are identical, every C component on every lane equals
// dot(dia[b,n], dd[b, m0+lane%16, n]).  Row norm^2 accumulated in VALU on
// the same loaded data, completed with shfl_xor(16).
// ---------------------------------------------------------------------------
__global__ void __launch_bounds__(256) k_avg_sim(const float* __restrict__ dia,
                                                 const float* __restrict__ dd,
                                                 const float* __restrict__ dnorm,
                                                 float* __restrict__ avg) {
    int blk  = blockIdx.x;          // 0..255
    int b    = blk >> 3;            // M_/16 = 8 tiles per batch
    int m0   = (blk & 7) << 4;
    int wave = threadIdx.x >> 5;    // 0..7
    int lane = threadIdx.x & 31;
    int mloc = lane & 15;
    int hi   = lane >> 4;           // K-half select

    const float* ddb  = dd  + (((size_t)b * M_ + (m0 + mloc)) * N_) * D_;
    const float* diab = dia + ((size_t)b * N_) * D_;

    float simsum = 0.f;
    float cnt    = 0.f;

    for (int ni = 0; ni < N_ / 8; ++ni) {
        int n = wave * (N_ / 8) + ni;
        const float* pdd = ddb  + (size_t)n * D_ + 2 * hi;
        const float* pda = diab + (size_t)n * D_ + 2 * hi;
        __builtin_prefetch(pdd + D_, 0, 1);   // next n-row of this wave

        v8f   c    = {0.f, 0.f, 0.f, 0.f, 0.f, 0.f, 0.f, 0.f};
        float nrm2 = 0.f;
        #pragma unroll 8
        for (int k = 0; k < D_; k += 4) {
            v2f bb = *(const v2f*)(pdd + k);   // dd row fragment
            v2f aa = *(const v2f*)(pda + k);   // dia fragment (bcast over M)
            nrm2 += bb.x * bb.x + bb.y * bb.y;
            c = __builtin_amdgcn_wmma_f32_16x16x4_f32(
                    false, aa, false, bb, (short)0, c, false, false);
        }
        nrm2 += __shfl_xor(nrm2, 16, 32);      // combine K halves
        float dot   = c[0];                    // dot for m = m0 + mloc
        float dn    = sqrtf(nrm2);
        float denom = fmaxf(dnorm[b * N_ + n] * dn, EPSQ);
        simsum += dot / denom;
        cnt    += (nrm2 > 0.f) ? 1.f : 0.f;    // non-padding node
    }

    __shared__ float s_sum[8][16];
    __shared__ float s_cnt[8][16];
    if (lane < 16) { s_sum[wave][mloc] = simsum; s_cnt[wave][mloc] = cnt; }
    __syncthreads();
    if (threadIdx.x < 16) {
        float ss = 0.f, cc = 0.f;
        #pragma unroll
        for (int w = 0; w < 8; ++w) { ss += s_sum[w][threadIdx.x]; cc += s_cnt[w][threadIdx.x]; }
        float ddnum = (cc == 0.f) ? NEG_BIG : cc;
        avg[b * M_ + m0 + threadIdx.x] = ss / ddnum;
    }
}

// ---------------------------------------------------------------------------
// K3: per-b max/argmax over M (first-index tie-break, like jnp.argmax)
// ---------------------------------------------------------------------------
__global__ void __launch_bounds__(128) k_argmax(const float* __restrict__ avg,
                                                float* __restrict__ vout,
                                                int* __restrict__ ixout) {
    int b = blockIdx.x;
    int t = threadIdx.x;             // 128 threads = M_
    float v  = avg[b * M_ + t];
    int   ix = t;
    #pragma unroll
    for (int off = 16; off >= 1; off >>= 1) {
        float ov = __shfl_xor(v, off, 32);
        int   oi = __shfl_xor(ix, off, 32);
        if (ov > v || (ov == v && oi < ix)) { v = ov; ix = oi; }
    }
    __shared__ float sv[4];
    __shared__ int   si[4];
    int w = t >> 5;
    if ((t & 31) == 0) { sv[w] = v; si[w] = ix; }
    __syncthreads();
    if (t == 0) {
        #pragma unroll
        for (int w2 = 1; w2 < 4; ++w2)
            if (sv[w2] > v || (sv[w2] == v && si[w2] < ix)) { v = sv[w2]; ix = si[w2]; }
        vout[b]  = v;
        ixout[b] = ix;
    }
}

// ---------------------------------------------------------------------------
// K4: out[b] = (v[b] > 0.5) ? dd[b, ix[b]] : dia[b]   (float4 copies)
// ---------------------------------------------------------------------------
__global__ void __launch_bounds__(256) k_output(const float* __restrict__ dia,
                                                const float* __restrict__ dd,
                                                const float* __restrict__ vws,
                                                const int* __restrict__ ixws,
                                                float* __restrict__ out) {
    int b     = blockIdx.x >> 4;            // 16 chunks per batch
    int chunk = blockIdx.x & 15;
    float v = vws[b];
    int  ix = ixws[b];
    size_t base = (size_t)b * N_ * D_;
    size_t off  = ((size_t)chunk * 256 + threadIdx.x) * 4;   // float4 units->floats
    const float* src = (v > 0.5f)
        ? dd + (((size_t)b * M_ + ix) * N_) * D_ + off
        : dia + base + off;
    *(float4*)(out + base + off) = *(const float4*)src;
}

// ---------------------------------------------------------------------------
extern "C" void kernel_launch(void* const* d_in, const int* in_sizes, int n_in,
                              void* d_out, int out_size, void* d_ws, size_t ws_size,
                              hipStream_t stream) {
    const float* dia = (const float*)d_in[0];   // [B, N, D]
    const float* dd  = (const float*)d_in[1];   // [B, M, N, D]
    float* out = (float*)d_out;                 // [B, N, D]

    float* ws    = (float*)d_ws;
    float* dnorm = ws;                          // B_*N_  = 2048 floats
    float* avg   = ws + B_ * N_;                // B_*M_  = 4096 floats
    float* vv    = avg + B_ * M_;               // B_ floats
    int*   ix    = (int*)(vv + B_);             // B_ ints

    k_dia_norm<<<(B_ * N_) / 8, 256, 0, stream>>>(dia, dnorm);
    k_avg_sim <<<B_ * (M_ / 16), 256, 0, stream>>>(dia, dd, dnorm, avg);
    k_argmax  <<<B_, 128, 0, stream>>>(avg, vv, ix);
    k_output  <<<B_ * 16, 256, 0, stream>>>(dia, dd, vv, ix, out);
}